// Encoder_21586505629729
// MI455X (gfx1250) — compile-verified
//
#include <hip/hip_runtime.h>

typedef __attribute__((ext_vector_type(2))) float v2f;
typedef __attribute__((ext_vector_type(8))) float v8f;

#define NB 8
#define NPTS 2048
#define KNN 20
#define GEPS 1e-5f
#define FNEG -3.402823466e38f

static __device__ __forceinline__ v8f wmma4(v2f a, v2f b, v8f c) {
  // D = A(16x4 f32) * B(4x16 f32) + C(16x16 f32)
  return __builtin_amdgcn_wmma_f32_16x16x4_f32(false, a, false, b, (short)0, c,
                                               false, false);
}

// ---------------------------------------------------------------- utility ---
__global__ void zero_kernel(float* __restrict__ p, int n) {
  int t = blockIdx.x * blockDim.x + threadIdx.x;
  if (t < n) p[t] = 0.f;
}

// xx[b][n] = sum_c x[b][c][n]^2
__global__ void colsq_kernel(const float* __restrict__ x, int cin, long bstride,
                             float* __restrict__ xx) {
  int t = blockIdx.x * blockDim.x + threadIdx.x;  // NB*NPTS
  int b = t / NPTS, n = t % NPTS;
  const float* xb = x + (long)b * bstride;
  float s = 0.f;
  for (int c = 0; c < cin; ++c) {
    float v = xb[(long)c * NPTS + n];
    s += v * v;
  }
  xx[t] = s;
}

// ------------------------------------------------------- Gram (WMMA GEMM) ---
// gram[i][j] = sum_c x[c][i]*x[c][j]   (one batch)
template <int CIN>
__global__ void gram_kernel(const float* __restrict__ x, float* __restrict__ gram) {
  const int tid = threadIdx.x;
  const int wid = tid >> 5, lane = tid & 31;
  const int hf = lane >> 4, l16 = lane & 15;
  const int m = blockIdx.y * 128 + wid * 16 + l16;  // row i
  const int n0 = blockIdx.x * 64;
  v8f acc[4] = {};
  for (int c0 = 0; c0 < CIN; c0 += 4) {
    const int ca = c0 + hf * 2;
    v2f a;
    a.x = (ca < CIN) ? x[(long)ca * NPTS + m] : 0.f;
    a.y = (ca + 1 < CIN) ? x[(long)(ca + 1) * NPTS + m] : 0.f;
#pragma unroll
    for (int s = 0; s < 4; ++s) {
      const int j = n0 + s * 16 + l16;
      v2f bb;
      bb.x = (ca < CIN) ? x[(long)ca * NPTS + j] : 0.f;
      bb.y = (ca + 1 < CIN) ? x[(long)(ca + 1) * NPTS + j] : 0.f;
      acc[s] = wmma4(a, bb, acc[s]);
    }
  }
  const int mrow = blockIdx.y * 128 + wid * 16 + hf * 8;
#pragma unroll
  for (int s = 0; s < 4; ++s) {
    const int j = n0 + s * 16 + l16;
#pragma unroll
    for (int r = 0; r < 8; ++r) gram[(long)(mrow + r) * NPTS + j] = acc[s][r];
  }
}

// ----------------------------------------------------------------- top-k ---
// score = 2*inner - xx[j]; 20 rounds of argmax (ties -> lower index, like lax.top_k)
__global__ void topk_kernel(const float* __restrict__ gram,
                            const float* __restrict__ xx, int* __restrict__ idxout) {
  __shared__ float sc[NPTS];
  __shared__ float rv[256];
  __shared__ int ri[256];
  const int i = blockIdx.x;
  const int tid = threadIdx.x;
  for (int j = tid; j < NPTS; j += 256)
    sc[j] = 2.f * gram[(long)i * NPTS + j] - xx[j];
  __syncthreads();
  for (int it = 0; it < KNN; ++it) {
    float bv = FNEG;
    int bi = NPTS;
    for (int j = tid; j < NPTS; j += 256) {
      float v = sc[j];
      if (v > bv || (v == bv && j < bi)) { bv = v; bi = j; }
    }
    rv[tid] = bv; ri[tid] = bi;
    __syncthreads();
    for (int off = 128; off > 0; off >>= 1) {
      if (tid < off) {
        float ov = rv[tid + off]; int oi = ri[tid + off];
        if (ov > rv[tid] || (ov == rv[tid] && oi < ri[tid])) { rv[tid] = ov; ri[tid] = oi; }
      }
      __syncthreads();
    }
    if (tid == 0) {
      idxout[i * KNN + it] = ri[0];
      sc[ri[0]] = FNEG;
    }
    __syncthreads();
  }
}

// ----------------------------------------- fused gather + edge conv (WMMA) ---
// f[o][p][k] = sum_{c<2CIN} W[o][c] * e[c][p][k],
//   e[c] = x[c][nbr]-x[c][ctr] (c<CIN) else x[c-CIN][ctr]
// STATS: accumulate per-channel sum/sumsq; FINAL: BN affine, max over k, lrelu.
template <int CIN, bool STATS>
__global__ void edge_conv_kernel(const float* __restrict__ x, long xbstride,
                                 const float* __restrict__ w,
                                 const int* __restrict__ idxg,
                                 float* __restrict__ ssum, float* __restrict__ ssq,
                                 const float* __restrict__ scl,
                                 const float* __restrict__ sft,
                                 float* __restrict__ out, long obstride, int ochan) {
  __shared__ int lidx[16 * KNN];
  __shared__ float fbuf[32][321];
  const int b = blockIdx.z;
  const int n0 = blockIdx.x * 16;
  const int tid = threadIdx.x;
  const int wid = tid >> 5, lane = tid & 31;
  const int hf = lane >> 4, l16 = lane & 15;
  const int ot = wid >> 2;   // o sub-tile (0..1)
  const int cg = wid & 3;    // column group (5 sub-tiles of 16 cols each)
  const int m0 = blockIdx.y * 32 + ot * 16;
  const float* xb = x + (long)b * xbstride;
  const int K2 = 2 * CIN;
  for (int t = tid; t < 16 * KNN; t += 256)
    lidx[t] = idxg[((long)b * NPTS + n0 + t / KNN) * KNN + t % KNN];
  __syncthreads();

  v8f acc[5] = {};
  for (int c0 = 0; c0 < K2; c0 += 4) {
    const int ca = c0 + hf * 2;
    v2f a;
    a.x = (ca < K2) ? w[(long)(m0 + l16) * K2 + ca] : 0.f;
    a.y = (ca + 1 < K2) ? w[(long)(m0 + l16) * K2 + ca + 1] : 0.f;
#pragma unroll
    for (int s = 0; s < 5; ++s) {
      const int col = (cg * 5 + s) * 16 + l16;
      const int p = col / KNN, kk = col % KNN;
      (void)kk;
      const int j = lidx[col];
      const int cn = n0 + p;
      v2f bb;
      {
        const int c = ca;
        float v;
        if (c < CIN) v = xb[(long)c * NPTS + j] - xb[(long)c * NPTS + cn];
        else if (c < K2) v = xb[(long)(c - CIN) * NPTS + cn];
        else v = 0.f;
        bb.x = v;
      }
      {
        const int c = ca + 1;
        float v;
        if (c < CIN) v = xb[(long)c * NPTS + j] - xb[(long)c * NPTS + cn];
        else if (c < K2) v = xb[(long)(c - CIN) * NPTS + cn];
        else v = 0.f;
        bb.y = v;
      }
      acc[s] = wmma4(a, bb, acc[s]);
    }
  }
  // dump D tiles to LDS: VGPR r holds M = r + 8*(lane/16), N = lane&15
#pragma unroll
  for (int s = 0; s < 5; ++s) {
    const int col = (cg * 5 + s) * 16 + l16;
#pragma unroll
    for (int r = 0; r < 8; ++r) fbuf[ot * 16 + hf * 8 + r][col] = acc[s][r];
  }
  __syncthreads();

  if (STATS) {
    const int row = tid >> 3, seg = tid & 7;  // 32 rows x 8 segments of 40 cols
    float s1 = 0.f, s2 = 0.f;
    for (int c = seg * 40; c < seg * 40 + 40; ++c) {
      float v = fbuf[row][c];
      s1 += v; s2 += v * v;
    }
    atomicAdd(&ssum[blockIdx.y * 32 + row], s1);
    atomicAdd(&ssq[blockIdx.y * 32 + row], s2);
  } else {
    for (int item = tid; item < 512; item += 256) {
      const int row = item >> 4, p = item & 15;
      const int o = blockIdx.y * 32 + row;
      const float sc = scl[o], sh = sft[o];
      float m = FNEG;
      for (int kk = 0; kk < KNN; ++kk)
        m = fmaxf(m, fbuf[row][p * KNN + kk] * sc + sh);
      m = (m >= 0.f) ? m : 0.2f * m;  // lrelu after max (monotone)
      out[(long)b * obstride + (long)(ochan + o) * NPTS + n0 + p] = m;
    }
  }
}

// BN finalize: biased variance over B*N*KNN
__global__ void bn_finalize_kernel(const float* __restrict__ ssum,
                                   const float* __restrict__ ssq,
                                   const float* __restrict__ g,
                                   const float* __restrict__ bb,
                                   float* __restrict__ scl, float* __restrict__ sft,
                                   int cout) {
  int o = blockIdx.x * blockDim.x + threadIdx.x;
  if (o >= cout) return;
  const float cnt = (float)NB * (float)NPTS * (float)KNN;
  float m = ssum[o] / cnt;
  float v = ssq[o] / cnt - m * m;
  float sc = g[o] * rsqrtf(v + GEPS);
  scl[o] = sc;
  sft[o] = bb[o] - m * sc;
}

// ------------------------------------------------- dense GEMM (WMMA, fp32) ---
// Y[b][o][n] = sum_c W[o][c] * X[b][c][n] + add[b][o]
template <int CIN>
__global__ void gemm_kernel(const float* __restrict__ w, int wstride,
                            const float* __restrict__ x, long xbstride,
                            const float* __restrict__ add, int addbstride,
                            float* __restrict__ y, long ybstride) {
  const int b = blockIdx.z;
  const int tid = threadIdx.x, wid = tid >> 5, lane = tid & 31;
  const int hf = lane >> 4, l16 = lane & 15;
  const int m0 = blockIdx.y * 128 + wid * 16;
  const int n0 = blockIdx.x * 64;
  const float* xb = x + (long)b * xbstride;
  v8f acc[4] = {};
  for (int c0 = 0; c0 < CIN; c0 += 4) {
    const int ca = c0 + hf * 2;
    const float* wp = w + (long)(m0 + l16) * wstride + ca;
    v2f a;
    a.x = wp[0];
    a.y = wp[1];
    if (c0 + 4 < CIN)
      __builtin_prefetch(xb + (long)(c0 + 4) * NPTS + n0 + l16, 0, 0);
#pragma unroll
    for (int s = 0; s < 4; ++s) {
      const int n = n0 + s * 16 + l16;
      v2f bb;
      bb.x = xb[(long)ca * NPTS + n];
      bb.y = xb[(long)(ca + 1) * NPTS + n];
      acc[s] = wmma4(a, bb, acc[s]);
    }
  }
  float* yb = y + (long)b * ybstride;
#pragma unroll
  for (int s = 0; s < 4; ++s) {
    const int n = n0 + s * 16 + l16;
#pragma unroll
    for (int r = 0; r < 8; ++r) {
      const int o = m0 + hf * 8 + r;
      float av = add ? add[b * addbstride + o] : 0.f;
      yb[(long)o * NPTS + n] = acc[s][r] + av;
    }
  }
}

// -------------------------------------------------------------- GroupNorm ---
__global__ void gn_stats_kernel(const float* __restrict__ y, int C,
                                float* __restrict__ gnm, float* __restrict__ gnr) {
  __shared__ float s1a[256], s2a[256];
  const int bg = blockIdx.x, b = bg >> 5, g = bg & 31;
  const int cpg = C / 32;
  const float* base = y + ((long)b * C + (long)g * cpg) * NPTS;
  const int total = cpg * NPTS;
  float s1 = 0.f, s2 = 0.f;
  for (int t = threadIdx.x; t < total; t += 256) {
    float v = base[t];
    s1 += v; s2 += v * v;
  }
  s1a[threadIdx.x] = s1; s2a[threadIdx.x] = s2;
  __syncthreads();
  for (int off = 128; off > 0; off >>= 1) {
    if (threadIdx.x < off) {
      s1a[threadIdx.x] += s1a[threadIdx.x + off];
      s2a[threadIdx.x] += s2a[threadIdx.x + off];
    }
    __syncthreads();
  }
  if (threadIdx.x == 0) {
    float m = s1a[0] / (float)total;
    float v = s2a[0] / (float)total - m * m;
    gnm[bg] = m;
    gnr[bg] = rsqrtf(v + GEPS);
  }
}

__global__ void gn_apply_kernel(float* __restrict__ y, int C,
                                const float* __restrict__ g, const float* __restrict__ bb,
                                const float* __restrict__ gnm,
                                const float* __restrict__ gnr, int act) {
  long t = (long)blockIdx.x * 256 + threadIdx.x;  // NB*C*NPTS
  int b = (int)(t / ((long)C * NPTS));
  long rem = t - (long)b * C * NPTS;
  int c = (int)(rem / NPTS);
  int grp = c / (C / 32);
  float m = gnm[b * 32 + grp], r = gnr[b * 32 + grp];
  float v = y[t];
  v = (v - m) * r * g[c] + bb[c];
  v = (act == 1) ? fmaxf(v, 0.f) : ((v >= 0.f) ? v : 0.2f * v);
  y[t] = v;
}

// ------------------------------------------------------------ global feats ---
__global__ void glb_kernel(const float* __restrict__ h, float* __restrict__ glb) {
  __shared__ float mx[256], sm[256];
  const int bc = blockIdx.x, b = bc >> 9, c = bc & 511;
  const float* row = h + ((long)b * 512 + c) * NPTS;
  float m = FNEG, s = 0.f;
  for (int n = threadIdx.x; n < NPTS; n += 256) {
    float v = row[n];
    m = fmaxf(m, v);
    s += v;
  }
  mx[threadIdx.x] = m; sm[threadIdx.x] = s;
  __syncthreads();
  for (int off = 128; off > 0; off >>= 1) {
    if (threadIdx.x < off) {
      mx[threadIdx.x] = fmaxf(mx[threadIdx.x], mx[threadIdx.x + off]);
      sm[threadIdx.x] += sm[threadIdx.x + off];
    }
    __syncthreads();
  }
  if (threadIdx.x == 0) {
    glb[b * 1024 + c] = mx[0];
    glb[b * 1024 + 512 + c] = sm[0] / (float)NPTS;
  }
}

// v[b][o] = bm1[o] + wm1[o][512:1536] . glb[b]
__global__ void vproj_kernel(const float* __restrict__ wm1,
                             const float* __restrict__ bm1,
                             const float* __restrict__ glb, float* __restrict__ v) {
  int t = blockIdx.x * 256 + threadIdx.x;  // NB*512
  int b = t >> 9, o = t & 511;
  float s = bm1[o];
  const float* wrow = wm1 + (long)o * 1536 + 512;
  const float* gb = glb + b * 1024;
  for (int c = 0; c < 1024; ++c) s += wrow[c] * gb[c];
  v[t] = s;
}

// --------------------------------------------------------- host orchestration
template <int CIN>
static void run_stage(const float* xin, long xbstride, const float* w,
                      const float* g, const float* bb, int cout, float* loc,
                      int ochan, float* gram, float* xx, int* idx, float* ssum,
                      float* ssq, float* scl, float* sft, hipStream_t stream) {
  colsq_kernel<<<NB * NPTS / 256, 256, 0, stream>>>(xin, CIN, xbstride, xx);
  for (int b = 0; b < NB; ++b) {
    gram_kernel<CIN><<<dim3(NPTS / 64, NPTS / 128), 256, 0, stream>>>(
        xin + (long)b * xbstride, gram);
    topk_kernel<<<NPTS, 256, 0, stream>>>(gram, xx + b * NPTS,
                                          idx + (long)b * NPTS * KNN);
  }
  zero_kernel<<<(cout + 255) / 256, 256, 0, stream>>>(ssum, cout);
  zero_kernel<<<(cout + 255) / 256, 256, 0, stream>>>(ssq, cout);
  dim3 eg(NPTS / 16, cout / 32, NB);
  edge_conv_kernel<CIN, true><<<eg, 256, 0, stream>>>(
      xin, xbstride, w, idx, ssum, ssq, nullptr, nullptr, nullptr, 0, 0);
  bn_finalize_kernel<<<(cout + 255) / 256, 256, 0, stream>>>(ssum, ssq, g, bb,
                                                             scl, sft, cout);
  edge_conv_kernel<CIN, false><<<eg, 256, 0, stream>>>(
      xin, xbstride, w, idx, nullptr, nullptr, scl, sft, loc, 512L * NPTS, ochan);
}

extern "C" void kernel_launch(void* const* d_in, const int* in_sizes, int n_in,
                              void* d_out, int out_size, void* d_ws, size_t ws_size,
                              hipStream_t stream) {
  (void)in_sizes; (void)n_in; (void)out_size; (void)ws_size;
  const float* xyz = (const float*)d_in[0];
  const float* w1 = (const float*)d_in[1];
  const float* g1 = (const float*)d_in[2];
  const float* b1 = (const float*)d_in[3];
  const float* w2 = (const float*)d_in[4];
  const float* g2 = (const float*)d_in[5];
  const float* b2 = (const float*)d_in[6];
  const float* w3 = (const float*)d_in[7];
  const float* g3 = (const float*)d_in[8];
  const float* b3 = (const float*)d_in[9];
  const float* w4 = (const float*)d_in[10];
  const float* g4 = (const float*)d_in[11];
  const float* b4 = (const float*)d_in[12];
  const float* w5 = (const float*)d_in[13];
  const float* g5 = (const float*)d_in[14];
  const float* b5 = (const float*)d_in[15];
  const float* wm1 = (const float*)d_in[16];
  const float* bm1 = (const float*)d_in[17];
  const float* g6 = (const float*)d_in[18];
  const float* b6 = (const float*)d_in[19];
  const float* wm2 = (const float*)d_in[20];
  const float* bm2 = (const float*)d_in[21];
  const float* g7 = (const float*)d_in[22];
  const float* b7 = (const float*)d_in[23];
  const float* wm3 = (const float*)d_in[24];
  const float* bm3 = (const float*)d_in[25];
  const float* g8 = (const float*)d_in[26];
  const float* b8 = (const float*)d_in[27];

  char* ws = (char*)d_ws;
  size_t off = 0;
  auto carve = [&](size_t bytes) {
    void* p = ws + off;
    off += (bytes + 255) & ~(size_t)255;
    return p;
  };
  float* loc  = (float*)carve((size_t)NB * 512 * NPTS * 4);  // x1..x4 slices
  float* h    = (float*)carve((size_t)NB * 512 * NPTS * 4);  // w5 out, then wm1 out
  float* h2   = (float*)carve((size_t)NB * 256 * NPTS * 4);  // wm2 out
  float* gram = (float*)carve((size_t)NPTS * NPTS * 4);      // per-batch reuse
  float* xx   = (float*)carve((size_t)NB * NPTS * 4);
  int*   idx  = (int*)carve((size_t)NB * NPTS * KNN * 4);
  float* ssum = (float*)carve(512 * 4);
  float* ssq  = (float*)carve(512 * 4);
  float* scl  = (float*)carve(512 * 4);
  float* sft  = (float*)carve(512 * 4);
  float* gnm  = (float*)carve(256 * 4);
  float* gnr  = (float*)carve(256 * 4);
  float* glb  = (float*)carve((size_t)NB * 1024 * 4);
  float* vb   = (float*)carve((size_t)NB * 512 * 4);

  // edge blocks -> loc channel slices (concat is free)
  run_stage<3>(xyz, 3L * NPTS, w1, g1, b1, 64, loc, 0, gram, xx, idx, ssum, ssq, scl, sft, stream);
  run_stage<64>(loc, 512L * NPTS, w2, g2, b2, 64, loc, 64, gram, xx, idx, ssum, ssq, scl, sft, stream);
  run_stage<64>(loc + 64L * NPTS, 512L * NPTS, w3, g3, b3, 128, loc, 128, gram, xx, idx, ssum, ssq, scl, sft, stream);
  run_stage<128>(loc + 128L * NPTS, 512L * NPTS, w4, g4, b4, 256, loc, 256, gram, xx, idx, ssum, ssq, scl, sft, stream);

  // h = lrelu(GN(w5 @ loc))
  gemm_kernel<512><<<dim3(32, 4, NB), 256, 0, stream>>>(w5, 512, loc, 512L * NPTS, nullptr, 0, h, 512L * NPTS);
  gn_stats_kernel<<<NB * 32, 256, 0, stream>>>(h, 512, gnm, gnr);
  gn_apply_kernel<<<NB * 512 * NPTS / 256, 256, 0, stream>>>(h, 512, g5, b5, gnm, gnr, 0);

  // glb = [max_n h ; mean_n h];  v = wm1[:,512:] @ glb + bm1
  glb_kernel<<<NB * 512, 256, 0, stream>>>(h, glb);
  vproj_kernel<<<NB * 512 / 256, 256, 0, stream>>>(wm1, bm1, glb, vb);

  // h = relu(GN(wm1[:, :512] @ loc + v))   (reuses h)
  gemm_kernel<512><<<dim3(32, 4, NB), 256, 0, stream>>>(wm1, 1536, loc, 512L * NPTS, vb, 512, h, 512L * NPTS);
  gn_stats_kernel<<<NB * 32, 256, 0, stream>>>(h, 512, gnm, gnr);
  gn_apply_kernel<<<NB * 512 * NPTS / 256, 256, 0, stream>>>(h, 512, g6, b6, gnm, gnr, 1);

  // h2 = relu(GN(wm2 @ h + bm2))
  gemm_kernel<512><<<dim3(32, 2, NB), 256, 0, stream>>>(wm2, 512, h, 512L * NPTS, bm2, 0, h2, 256L * NPTS);
  gn_stats_kernel<<<NB * 32, 256, 0, stream>>>(h2, 256, gnm, gnr);
  gn_apply_kernel<<<NB * 256 * NPTS / 256, 256, 0, stream>>>(h2, 256, g7, b7, gnm, gnr, 1);

  // out = relu(GN(wm3 @ h2 + bm3))
  float* out = (float*)d_out;
  gemm_kernel<256><<<dim3(32, 1, NB), 256, 0, stream>>>(wm3, 256, h2, 256L * NPTS, bm3, 0, out, 128L * NPTS);
  gn_stats_kernel<<<NB * 32, 256, 0, stream>>>(out, 128, gnm, gnr);
  gn_apply_kernel<<<NB * 128 * NPTS / 256, 256, 0, stream>>>(out, 128, g8, b8, gnm, gnr, 1);
}